// SelectiveSSM_74337293959200
// MI455X (gfx1250) — compile-verified
//
#include <hip/hip_runtime.h>
#include <hip/hip_bf16.h>
#include <math.h>

// ---------------------------------------------------------------------------
// Selective SSM (Mamba) forward for MI455X / gfx1250 (wave32, WMMA).
//   proj = x^T @ W_xproj^T                 (M=4096, N=96,   K=1024) f32 WMMA
//   dt   = softplus(dt_in @ W_dt^T + b_dt) (M=4096, N=1024, K=64)  f32 WMMA
//   scan = sequential over L, 16 lanes per (b,d) channel (one lane per state)
// N-register-blocked: the A fragment (large tensor) is loaded once per k-step
// and reused across all N tiles of the wave, cutting A traffic 6x / 4x.
// ---------------------------------------------------------------------------

typedef __attribute__((ext_vector_type(2))) float v2f;
typedef __attribute__((ext_vector_type(8))) float v8f;

#define D_MODEL 1024
#define D_STATE 16
#define DT_RANK 64
#define BATCH   2
#define SEQLEN  2048
#define P_DIM   (DT_RANK + 2 * D_STATE)   // 96
#define M_TOT   (BATCH * SEQLEN)          // 4096

// ---------------------------------------------------------------------------
// GEMM1: proj[m][p] = sum_k x[b, k, l] * W_xproj[p, k],  m = b*L + l
// One wave computes a full 16(M) x 96(P) strip: 6 accumulators, A reused 6x.
// A-frag layout (ISA 7.12.2, 32-bit A 16x4): lanes 0-15 -> K = k0,k0+1,
// lanes 16-31 -> K = k0+2,k0+3; M = lane&15.  B-frag mirrors (N = lane&15).
// C/D: VGPR r holds row m0 + r + 8*(lane>=16), column (lane&15).
// ---------------------------------------------------------------------------
__global__ __launch_bounds__(256) void proj_gemm_kernel(
    const float* __restrict__ x, const float* __restrict__ Wxp,
    float* __restrict__ proj) {
  const int lane = threadIdx.x & 31;
  const int wave = threadIdx.x >> 5;
  const int W    = blockIdx.x * 8 + wave;   // one wave per 16-row M strip
  const int m0   = W * 16;
  const int half = lane >> 4;
  const int lid  = lane & 15;

  // A row for this lane: m = m0 + lid -> (b, l). SEQLEN % 16 == 0 so a strip
  // never crosses the batch boundary.
  const int mA = m0 + lid;
  const int b  = mA / SEQLEN;
  const int l  = mA % SEQLEN;
  const float* __restrict__ xb = x + (size_t)b * D_MODEL * SEQLEN + l; // [k] stride SEQLEN
  const float* __restrict__ wp = Wxp + (size_t)lid * D_MODEL;          // row lid, + t*16 rows

  v8f acc[6];
#pragma unroll
  for (int t = 0; t < 6; ++t) acc[t] = (v8f){};

#pragma unroll 2
  for (int k0 = 0; k0 < D_MODEL; k0 += 4) {
    const int kA = k0 + 2 * half;
    v2f a;
    a[0] = xb[(size_t)(kA + 0) * SEQLEN];
    a[1] = xb[(size_t)(kA + 1) * SEQLEN];
#pragma unroll
    for (int t = 0; t < 6; ++t) {
      const float* __restrict__ w = wp + (size_t)t * 16 * D_MODEL;
      v2f bf;
      bf[0] = w[kA + 0];
      bf[1] = w[kA + 1];
      acc[t] = __builtin_amdgcn_wmma_f32_16x16x4_f32(
          false, a, false, bf, (short)0, acc[t], false, false);
    }
  }

#pragma unroll
  for (int t = 0; t < 6; ++t) {
#pragma unroll
    for (int r = 0; r < 8; ++r) {
      const int m = m0 + r + 8 * half;
      proj[(size_t)m * P_DIM + t * 16 + lid] = acc[t][r];
    }
  }
}

// ---------------------------------------------------------------------------
// GEMM2: dt[m][d] = softplus( sum_r proj[m][r] * W_dt[d][r] + b_dt[d] )
// M=4096, N=1024, K=64.  One wave computes 16(M) x 64(N): 4 accumulators,
// A (proj) fragment reused 4x.
// ---------------------------------------------------------------------------
__global__ __launch_bounds__(256) void dt_gemm_kernel(
    const float* __restrict__ proj, const float* __restrict__ Wdt,
    const float* __restrict__ bdt, float* __restrict__ dt) {
  const int lane = threadIdx.x & 31;
  const int wave = threadIdx.x >> 5;
  const int W    = blockIdx.x * 8 + wave;
  const int NG   = D_MODEL / 64;            // 16 N-groups of 64
  const int m0   = (W / NG) * 16;
  const int n0   = (W % NG) * 64;
  const int half = lane >> 4;
  const int lid  = lane & 15;

  const float* __restrict__ ap = proj + (size_t)(m0 + lid) * P_DIM;   // dt_in rows
  const float* __restrict__ wp = Wdt + (size_t)(n0 + lid) * DT_RANK;  // W_dt rows

  v8f acc[4];
#pragma unroll
  for (int t = 0; t < 4; ++t) acc[t] = (v8f){};

#pragma unroll
  for (int k0 = 0; k0 < DT_RANK; k0 += 4) {
    const int kA = k0 + 2 * half;
    v2f a;
    a[0] = ap[kA + 0];
    a[1] = ap[kA + 1];
#pragma unroll
    for (int t = 0; t < 4; ++t) {
      const float* __restrict__ w = wp + (size_t)t * 16 * DT_RANK;
      v2f bf;
      bf[0] = w[kA + 0];
      bf[1] = w[kA + 1];
      acc[t] = __builtin_amdgcn_wmma_f32_16x16x4_f32(
          false, a, false, bf, (short)0, acc[t], false, false);
    }
  }

#pragma unroll
  for (int t = 0; t < 4; ++t) {
    const int   n    = n0 + t * 16 + lid;
    const float bias = bdt[n];
#pragma unroll
    for (int r = 0; r < 8; ++r) {
      const int   m  = m0 + r + 8 * half;
      const float z  = acc[t][r] + bias;
      const float sp = (z > 20.0f) ? z : __logf(1.0f + __expf(z));
      dt[(size_t)m * D_MODEL + n] = sp;
    }
  }
}

// ---------------------------------------------------------------------------
// Scan: 16 lanes per (b,d) channel, one lane per state n.
//   h_n <- exp(dt * A_n) * h_n + dt * B_n * x ;  y = sum_n h_n * C_n + D*x
// Cross-lane sum over the 16-lane group via shfl_xor (wave32).
// deltaA / deltaB_x are never materialized (saves ~536 MB of traffic vs ref).
// 2048 channels * 16 lanes = 1024 wave32s of parallelism.
// ---------------------------------------------------------------------------
__global__ __launch_bounds__(256) void ssm_scan_kernel(
    const float* __restrict__ x, const float* __restrict__ proj,
    const float* __restrict__ dt, const float* __restrict__ A_log,
    const float* __restrict__ Dp, float* __restrict__ out) {
  const int tid = threadIdx.x;
  const int g   = blockIdx.x * 16 + (tid >> 4);   // channel index (b,d)
  const int n   = tid & 15;                       // state index
  const int b   = g / D_MODEL;
  const int d   = g % D_MODEL;

  const float An = -__expf(A_log[d * D_STATE + n]);  // A = -exp(A_log)
  const float Dd = Dp[d];

  const float* __restrict__ xr  = x + ((size_t)b * D_MODEL + d) * SEQLEN;
  const float* __restrict__ dtr = dt + (size_t)b * SEQLEN * D_MODEL + d;
  const float* __restrict__ pr  = proj + (size_t)b * SEQLEN * P_DIM;
  float* __restrict__ yr        = out + ((size_t)b * D_MODEL + d) * SEQLEN;

  float h = 0.0f;
  for (int l = 0; l < SEQLEN; ++l) {
    const float dtv = dtr[(size_t)l * D_MODEL];
    const float xv  = xr[l];
    const float Bn  = pr[l * P_DIM + DT_RANK + n];
    const float Cn  = pr[l * P_DIM + DT_RANK + D_STATE + n];

    const float a = __expf(dtv * An);
    h = a * h + (dtv * xv) * Bn;

    float p = h * Cn;
    p += __shfl_xor(p, 1);
    p += __shfl_xor(p, 2);
    p += __shfl_xor(p, 4);
    p += __shfl_xor(p, 8);

    if (n == 0) yr[l] = p + Dd * xv;
  }
}

// ---------------------------------------------------------------------------
extern "C" void kernel_launch(void* const* d_in, const int* in_sizes, int n_in,
                              void* d_out, int out_size, void* d_ws, size_t ws_size,
                              hipStream_t stream) {
  const float* x     = (const float*)d_in[0];
  const float* Wxp   = (const float*)d_in[1];
  const float* Wdt   = (const float*)d_in[2];
  const float* bdt   = (const float*)d_in[3];
  const float* A_log = (const float*)d_in[4];
  const float* Dp    = (const float*)d_in[5];
  float* out = (float*)d_out;

  // Workspace layout: proj (4096*96 f32 = 1.57 MB) | dt (4096*1024 f32 = 16.8 MB)
  float* proj = (float*)d_ws;
  float* dtb  = proj + (size_t)M_TOT * P_DIM;

  // GEMM1: 256 M-strips (16x96 each) = 256 waves, 8 waves/block -> 32 blocks
  proj_gemm_kernel<<<(M_TOT / 16) / 8, 256, 0, stream>>>(x, Wxp, proj);
  // GEMM2: 256 M-tiles x 16 N-groups = 4096 waves -> 512 blocks
  dt_gemm_kernel<<<(M_TOT / 16) * (D_MODEL / 64) / 8, 256, 0, stream>>>(proj, Wdt, bdt, dtb);
  // Scan: 2048 channels x 16 lanes = 1024 waves, 16 channels/block
  ssm_scan_kernel<<<(BATCH * D_MODEL) / 16, 256, 0, stream>>>(x, proj, dtb, A_log, Dp, out);
}